// DifferentiableDLT_74217034875425
// MI455X (gfx1250) — compile-verified
//
#include <hip/hip_runtime.h>
#include <math.h>

// DifferentiableDLT for MI455X (gfx1250, wave32).
// One workgroup per batch image. 1024 grid points staged in LDS,
// normal equations G^T G accumulated with V_WMMA_F32_16X16X4_F32.

typedef __attribute__((ext_vector_type(2))) float v2f;
typedef __attribute__((ext_vector_type(8))) float v8f;

#define NPTS   1024
#define NGRID  32
#define HF     320
#define WF     576
#define BLOCK  256
#define NWAVES 8          // 256 threads / wave32
#define EPS_L  1e-6f

__device__ __forceinline__ float wave_reduce_sum(float v) {
  // wave32 butterfly reduction
  v += __shfl_xor(v, 16, 32);
  v += __shfl_xor(v, 8, 32);
  v += __shfl_xor(v, 4, 32);
  v += __shfl_xor(v, 2, 32);
  v += __shfl_xor(v, 1, 32);
  return v;
}

__device__ __forceinline__ void matmul3(const float A[3][3], const float B[3][3],
                                        float C[3][3]) {
  for (int i = 0; i < 3; ++i)
    for (int j = 0; j < 3; ++j)
      C[i][j] = A[i][0] * B[0][j] + A[i][1] * B[1][j] + A[i][2] * B[2][j];
}

__global__ __launch_bounds__(BLOCK)
void dlt_kernel(const float* __restrict__ flow, const float* __restrict__ mask,
                const int* __restrict__ p_img_h, const int* __restrict__ p_img_w,
                float* __restrict__ out) {
  __shared__ float sX[NPTS], sY[NPTS], dXs[NPTS], dYs[NPTS], sW[NPTS];
  __shared__ float sD[NWAVES][16][16];   // per-wave partial G^T G
  __shared__ float sAtG[16][16];         // reduced G^T G
  __shared__ float sPart[NWAVES][8];
  __shared__ float sStat[12];

  const int b    = blockIdx.x;
  const int t    = threadIdx.x;
  const int lane = t & 31;
  const int wave = t >> 5;

  const float img_h = (float)p_img_h[0];
  const float img_w = (float)p_img_w[0];
  const float scl_x = (img_w - 1.0f) / (float)(WF - 1);
  const float scl_y = (img_h - 1.0f) / (float)(HF - 1);

  // grid: linspace over [margin, dim-margin-1], 32 samples
  const int mh = (int)((float)HF * 0.05f);   // 16
  const int mw = (int)((float)WF * 0.05f);   // 28
  const float x_start = (float)mw;
  const float x_step  = ((float)(WF - mw - 1) - x_start) / (float)(NGRID - 1);
  const float y_start = (float)mh;
  const float y_step  = ((float)(HF - mh - 1) - y_start) / (float)(NGRID - 1);

  const float* fx_base = flow + (size_t)b * 2 * HF * WF;
  const float* fy_base = fx_base + (size_t)HF * WF;
  const float* m_base  = mask + (size_t)b * HF * WF;

  // ---- Stage A: bilinear sample 4 points/thread, stage to LDS ----
  float lsx = 0.f, lsy = 0.f, ldx = 0.f, ldy = 0.f, lw = 0.f;
  for (int i = 0; i < NPTS / BLOCK; ++i) {
    int p  = t + i * BLOCK;
    int iy = p >> 5, ix = p & 31;
    float x = x_start + (float)ix * x_step;
    float y = y_start + (float)iy * y_step;
    float fx0 = floorf(x), fy0 = floorf(y);
    float wx = x - fx0, wy = y - fy0;
    int x0i = min(max((int)fx0, 0), WF - 1);
    int x1i = min(x0i + 1, WF - 1);
    int y0i = min(max((int)fy0, 0), HF - 1);
    int y1i = min(y0i + 1, HF - 1);
    float w00 = (1.f - wx) * (1.f - wy), w01 = wx * (1.f - wy);
    float w10 = (1.f - wx) * wy,         w11 = wx * wy;
    int i00 = y0i * WF + x0i, i01 = y0i * WF + x1i;
    int i10 = y1i * WF + x0i, i11 = y1i * WF + x1i;
    float fu = fx_base[i00] * w00 + fx_base[i01] * w01 +
               fx_base[i10] * w10 + fx_base[i11] * w11;
    float fv = fy_base[i00] * w00 + fy_base[i01] * w01 +
               fy_base[i10] * w10 + fy_base[i11] * w11;
    float w  = m_base[i00] * w00 + m_base[i01] * w01 +
               m_base[i10] * w10 + m_base[i11] * w11;
    w = fmaxf(w, 0.0f);
    float sx = x * scl_x,        sy = y * scl_y;
    float dx = (x + fu) * scl_x, dy = (y + fv) * scl_y;
    sX[p] = sx; sY[p] = sy; dXs[p] = dx; dYs[p] = dy; sW[p] = w;
    lsx += sx; lsy += sy; ldx += dx; ldy += dy; lw += w;
  }

  // ---- Stage B: means + support ----
  lsx = wave_reduce_sum(lsx); lsy = wave_reduce_sum(lsy);
  ldx = wave_reduce_sum(ldx); ldy = wave_reduce_sum(ldy);
  lw  = wave_reduce_sum(lw);
  if (lane == 0) {
    sPart[wave][0] = lsx; sPart[wave][1] = lsy;
    sPart[wave][2] = ldx; sPart[wave][3] = ldy;
    sPart[wave][4] = lw;
  }
  __syncthreads();
  if (t == 0) {
    float a0 = 0, a1 = 0, a2 = 0, a3 = 0, a4 = 0;
    for (int wv = 0; wv < NWAVES; ++wv) {
      a0 += sPart[wv][0]; a1 += sPart[wv][1]; a2 += sPart[wv][2];
      a3 += sPart[wv][3]; a4 += sPart[wv][4];
    }
    sStat[0] = a0 / (float)NPTS;  // mean src x
    sStat[1] = a1 / (float)NPTS;  // mean src y
    sStat[2] = a2 / (float)NPTS;  // mean dst x
    sStat[3] = a3 / (float)NPTS;  // mean dst y
    sStat[4] = a4 / (float)NPTS;  // support
  }
  __syncthreads();
  const float mxs = sStat[0], mys = sStat[1], mxd = sStat[2], myd = sStat[3];

  // ---- Stage C: Hartley scales (mean of norms / sqrt(2)) ----
  float lns = 0.f, lnd = 0.f;
  for (int i = 0; i < NPTS / BLOCK; ++i) {
    int p = t + i * BLOCK;
    float cx = sX[p] - mxs, cy = sY[p] - mys;
    lns += sqrtf(cx * cx + cy * cy);
    cx = dXs[p] - mxd; cy = dYs[p] - myd;
    lnd += sqrtf(cx * cx + cy * cy);
  }
  lns = wave_reduce_sum(lns); lnd = wave_reduce_sum(lnd);
  if (lane == 0) { sPart[wave][5] = lns; sPart[wave][6] = lnd; }
  __syncthreads();
  if (t == 0) {
    float s0 = 0, s1 = 0;
    for (int wv = 0; wv < NWAVES; ++wv) { s0 += sPart[wv][5]; s1 += sPart[wv][6]; }
    float rsqrt2 = 0.70710678118654752440f;
    float ss = fmaxf((s0 / (float)NPTS) * rsqrt2, 1e-8f);
    float sd = fmaxf((s1 / (float)NPTS) * rsqrt2, 1e-8f);
    sStat[5] = 1.0f / ss;   // inv scale src
    sStat[6] = 1.0f / sd;   // inv scale dst
    sStat[7] = ss;
    sStat[8] = sd;
  }
  __syncthreads();
  const float invs = sStat[5], invd = sStat[6];

  // normalize points in place; w -> sqrt(w)
  for (int i = 0; i < NPTS / BLOCK; ++i) {
    int p = t + i * BLOCK;
    sX[p]  = (sX[p]  - mxs) * invs;
    sY[p]  = (sY[p]  - mys) * invs;
    dXs[p] = (dXs[p] - mxd) * invd;
    dYs[p] = (dYs[p] - myd) * invd;
    sW[p]  = sqrtf(sW[p]);
  }
  __syncthreads();

  // ---- Stage D: accumulate G^T G (G = [A | b], 2048 x 9, weighted) ----
  // Each wave: 128 points -> 64 rank-4 updates, 2 points (4 rows) per WMMA.
  // Operand layout for both A(16x4) and B(4x16) f32 coincides per lane:
  //   e = lane&15 (element index), half = lane>>4 selects point of the pair,
  //   v2f = { rw*g_row1[e], rw*g_row2[e] }.
  //
  // g_row1 = [x, y, 1, 0, 0, 0, -xp*x, -xp*y, xp, 0...]
  // g_row2 = [0, 0, 0, x, y, 1, -yp*x, -yp*y, yp, 0...]
  //
  // Branchless: per-lane 0/1 coefficient flags hoisted out of the loop so the
  // body is pure FMA + ds_load + v_wmma (no divergent switch).
  {
    const int e     = lane & 15;
    const int half  = lane >> 4;
    const int pbase = wave * (NPTS / NWAVES);

    const float a_x  = (e == 0) ? 1.0f : 0.0f;
    const float a_y  = (e == 1) ? 1.0f : 0.0f;
    const float a_c  = (e == 2) ? 1.0f : 0.0f;
    const float a_px = (e == 6) ? 1.0f : 0.0f;
    const float a_py = (e == 7) ? 1.0f : 0.0f;
    const float a_b  = (e == 8) ? 1.0f : 0.0f;
    const float b_x  = (e == 3) ? 1.0f : 0.0f;
    const float b_y  = (e == 4) ? 1.0f : 0.0f;
    const float b_c  = (e == 5) ? 1.0f : 0.0f;
    const float b_px = (e == 6) ? 1.0f : 0.0f;
    const float b_py = (e == 7) ? 1.0f : 0.0f;
    const float b_b  = (e == 8) ? 1.0f : 0.0f;

#if defined(__gfx1250__) && __has_builtin(__builtin_amdgcn_wmma_f32_16x16x4_f32)
    v8f acc0 = {};
    v8f acc1 = {};
    for (int it = 0; it < (NPTS / NWAVES) / 2; it += 2) {
      // --- pair 0 ---
      {
        int p = pbase + it * 2 + half;
        float x = sX[p], y = sY[p], xp = dXs[p], yp = dYs[p], rw = sW[p];
        float xpx = xp * x, xpy = xp * y, ypx = yp * x, ypy = yp * y;
        v2f ab;
        ab.x = rw * fmaf(a_x, x, fmaf(a_y, y, fmaf(a_px, -xpx,
                     fmaf(a_py, -xpy, fmaf(a_b, xp, a_c)))));
        ab.y = rw * fmaf(b_x, x, fmaf(b_y, y, fmaf(b_px, -ypx,
                     fmaf(b_py, -ypy, fmaf(b_b, yp, b_c)))));
        acc0 = __builtin_amdgcn_wmma_f32_16x16x4_f32(
            false, ab, false, ab, (short)0, acc0, false, false);
      }
      // --- pair 1 (independent accumulator for ILP) ---
      {
        int p = pbase + (it + 1) * 2 + half;
        float x = sX[p], y = sY[p], xp = dXs[p], yp = dYs[p], rw = sW[p];
        float xpx = xp * x, xpy = xp * y, ypx = yp * x, ypy = yp * y;
        v2f ab;
        ab.x = rw * fmaf(a_x, x, fmaf(a_y, y, fmaf(a_px, -xpx,
                     fmaf(a_py, -xpy, fmaf(a_b, xp, a_c)))));
        ab.y = rw * fmaf(b_x, x, fmaf(b_y, y, fmaf(b_px, -ypx,
                     fmaf(b_py, -ypy, fmaf(b_b, yp, b_c)))));
        acc1 = __builtin_amdgcn_wmma_f32_16x16x4_f32(
            false, ab, false, ab, (short)0, acc1, false, false);
      }
    }
    for (int r = 0; r < 8; ++r)
      sD[wave][r + 8 * half][e] = acc0[r] + acc1[r];
#else
    // scalar fallback with identical D layout
    float acc[8] = {0, 0, 0, 0, 0, 0, 0, 0};
    for (int it = 0; it < (NPTS / NWAVES) / 2; ++it) {
      for (int k = 0; k < 4; ++k) {
        int p = pbase + it * 2 + (k >> 1);
        int row = k & 1;
        float x = sX[p], y = sY[p], xp = dXs[p], yp = dYs[p], rw = sW[p];
        float xpx = xp * x, xpy = xp * y, ypx = yp * x, ypy = yp * y;
        float gb;
        if (row == 0)
          gb = rw * (a_x * x + a_y * y + a_c - a_px * xpx - a_py * xpy + a_b * xp);
        else
          gb = rw * (b_x * x + b_y * y + b_c - b_px * ypx - b_py * ypy + b_b * yp);
        for (int r = 0; r < 8; ++r) {
          int M = r + 8 * half;
          float ga;
          float m_x  = (M == 0), m_y  = (M == 1), m_c  = (M == 2);
          float m_px = (M == 6), m_py = (M == 7), m_b  = (M == 8);
          float n_x  = (M == 3), n_y  = (M == 4), n_c  = (M == 5);
          if (row == 0)
            ga = rw * (m_x * x + m_y * y + m_c - m_px * xpx - m_py * xpy + m_b * xp);
          else
            ga = rw * (n_x * x + n_y * y + n_c - m_px * ypx - m_py * ypy + m_b * yp);
          acc[r] += ga * gb;
        }
      }
    }
    for (int r = 0; r < 8; ++r) sD[wave][r + 8 * half][e] = acc[r];
#endif
  }
  __syncthreads();

  // reduce the 8 wave partials (deterministic): thread t owns one (M,N)
  {
    int M = t >> 4, N = t & 15;
    float s = 0.f;
    for (int wv = 0; wv < NWAVES; ++wv) s += sD[wv][M][N];
    sAtG[M][N] = s;
  }
  __syncthreads();

  // ---- Stage E: solve + denormalize + write (thread 0) ----
  if (t == 0) {
    // (AtA + eps*I) h = Atb ; AtA = sAtG[0..7][0..7], Atb = sAtG[i][8]
    float M9[8][9];
    for (int i = 0; i < 8; ++i) {
      for (int j = 0; j < 8; ++j)
        M9[i][j] = sAtG[i][j] + (i == j ? EPS_L : 0.0f);
      M9[i][8] = sAtG[i][8];
    }
    // Gauss-Jordan with partial pivoting
    for (int c = 0; c < 8; ++c) {
      int piv = c; float best = fabsf(M9[c][c]);
      for (int r = c + 1; r < 8; ++r) {
        float v = fabsf(M9[r][c]);
        if (v > best) { best = v; piv = r; }
      }
      if (piv != c)
        for (int j = c; j < 9; ++j) {
          float tmp = M9[c][j]; M9[c][j] = M9[piv][j]; M9[piv][j] = tmp;
        }
      float inv = 1.0f / M9[c][c];
      for (int r = 0; r < 8; ++r) {
        if (r == c) continue;
        float f = M9[r][c] * inv;
        for (int j = c; j < 9; ++j) M9[r][j] -= f * M9[c][j];
      }
    }
    float h[8];
    for (int i = 0; i < 8; ++i) h[i] = M9[i][8] / M9[i][i];

    float Hn[3][3] = {{h[0], h[1], h[2]}, {h[3], h[4], h[5]}, {h[6], h[7], 1.0f}};
    float sd = sStat[8];
    float Ts[3][3]  = {{invs, 0.f, -mxs * invs}, {0.f, invs, -mys * invs}, {0.f, 0.f, 1.f}};
    float Tdi[3][3] = {{sd, 0.f, mxd}, {0.f, sd, myd}, {0.f, 0.f, 1.f}};
    float Tmp[3][3], H[3][3];
    matmul3(Tdi, Hn, Tmp);
    matmul3(Tmp, Ts, H);

    float s22 = H[2][2];
    float sgn = (s22 < 0.0f) ? -1.0f : 1.0f;
    float sdiv = sgn * fmaxf(fabsf(s22), 1e-8f);
    float inv_sdiv = 1.0f / sdiv;
    bool ok = sStat[4] > 0.0001f;

    float* o = out + (size_t)b * 9;
    for (int i = 0; i < 3; ++i)
      for (int j = 0; j < 3; ++j) {
        float v = H[i][j] * inv_sdiv;
        if (!ok) v = (i == j) ? 1.0f : 0.0f;
        o[i * 3 + j] = v;
      }
  }
}

extern "C" void kernel_launch(void* const* d_in, const int* in_sizes, int n_in,
                              void* d_out, int out_size, void* d_ws, size_t ws_size,
                              hipStream_t stream) {
  (void)in_sizes; (void)n_in; (void)d_ws; (void)ws_size;
  const float* flow  = (const float*)d_in[0];
  const float* mask  = (const float*)d_in[1];
  const int*   img_h = (const int*)d_in[2];
  const int*   img_w = (const int*)d_in[3];
  float* out = (float*)d_out;
  int B = out_size / 9;   // 64 for the reference setup
  hipLaunchKernelGGL(dlt_kernel, dim3(B), dim3(BLOCK), 0, stream,
                     flow, mask, img_h, img_w, out);
}